// BOW_model2_5798205849946
// MI455X (gfx1250) — compile-verified
//
#include <hip/hip_runtime.h>
#include <hip/hip_bf16.h>
#include <math.h>

// ---------------------------------------------------------------------------
// BOW model: gather/segment-mean -> 3x (Linear + BatchNorm(train) + ReLU)
//            -> Linear(256->1) -> BCEWithLogits mean loss.
// GEMMs use CDNA5 V_WMMA_F32_16X16X4_F32 (full f32 precision; GEMMs are not
// the bottleneck -- the 512MB embedding gather is, and the 100MB table lives
// in the 192MB L2, so the gather kernel is optimized for L2 bandwidth and
// VMEM issue rate with b128 bursts).
// ---------------------------------------------------------------------------

typedef float v2f __attribute__((ext_vector_type(2)));
typedef float v8f __attribute__((ext_vector_type(8)));

#define VOCAB   100000
#define H       256
#define B_SAMP  4096
#define L_TOK   128
#define BN_EPS  1e-5f

// ---------------------------------------------------------------------------
// 1) Gather + segment mean.  One block (256 threads) per sample.
//    Lanes are split into 4 sub-groups of 64; each sub-group streams one
//    embedding row per step as float4 (global_load_b128, 1KB per row,
//    512B per wave burst), processing 4 tokens concurrently.  A small LDS
//    reduction folds the 4 sub-group partials.
// ---------------------------------------------------------------------------
__global__ __launch_bounds__(256)
void k_gather_mean(const int* __restrict__ tok,
                   const float* __restrict__ emb,
                   float* __restrict__ bow) {
    const int b = blockIdx.x;          // sample
    const int t = threadIdx.x;
    __shared__ int    ids[L_TOK];
    __shared__ float4 part[4][H / 4];

    if (t < L_TOK) ids[t] = tok[b * L_TOK + t];
    __syncthreads();

    const int sub = t >> 6;            // token sub-group 0..3
    const int f   = t & 63;            // float4 index within the row

    float4 s = make_float4(0.f, 0.f, 0.f, 0.f);
    #pragma unroll 4
    for (int i = sub; i < L_TOK; i += 4) {
        const float4 v = ((const float4*)(emb + (size_t)ids[i] * H))[f];
        s.x += v.x; s.y += v.y; s.z += v.z; s.w += v.w;
    }
    part[sub][f] = s;
    __syncthreads();

    if (sub == 0) {
        const float4 a = part[0][f], c = part[1][f],
                     d = part[2][f], e = part[3][f];
        float4 r;
        r.x = (a.x + c.x + d.x + e.x) * (1.0f / (float)L_TOK);
        r.y = (a.y + c.y + d.y + e.y) * (1.0f / (float)L_TOK);
        r.z = (a.z + c.z + d.z + e.z) * (1.0f / (float)L_TOK);
        r.w = (a.w + c.w + d.w + e.w) * (1.0f / (float)L_TOK);
        ((float4*)(bow + (size_t)b * H))[f] = r;
    }
}

// ---------------------------------------------------------------------------
// 2) GEMM  Y[M,256] = act(X)[M,256] @ W[256,256] + bias
//    act(x) = APPLY_BN ? relu(x*scale[c] + shift[c]) : x  (c = K column of X)
//    APPLY_BN is a template parameter so the K-loop has no control flow.
//    Block = 256 threads = 8 waves; each wave owns a 16x16 tile via
//    V_WMMA_F32_16X16X4_F32; block tile = 64 rows x 32 cols; grid = 512.
//    A frag (16x4 f32): lane l: m = l&15, VGPR j -> K = k0 + 2*(l>>4) + j
//    B frag (4x16 f32): lane l: n = l&15, VGPR j -> K = k0 + 2*(l>>4) + j
//    C/D (16x16 f32):   lane l: n = l&15, VGPR i -> M = i + 8*(l>>4)
// ---------------------------------------------------------------------------
template <int APPLY_BN>
__global__ __launch_bounds__(256)
void k_gemm_bn(const float* __restrict__ X,
               const float* __restrict__ W,
               const float* __restrict__ bias,
               float* __restrict__ Y,
               const float* __restrict__ ss) {  // [0:256)=scale, [256:512)=shift
    __shared__ float s_sc[H];
    __shared__ float s_sh[H];
    const int tid = threadIdx.x;
    if (APPLY_BN) {
        s_sc[tid] = ss[tid];
        s_sh[tid] = ss[H + tid];
        __syncthreads();
    }

    const int wave = tid >> 5;           // 0..7
    const int lane = tid & 31;
    const int p    = lane & 15;
    const int hi   = lane >> 4;          // 0 or 1

    const int bm = blockIdx.x >> 3;      // 64 row-blocks
    const int bn = blockIdx.x & 7;       // 8 col-blocks
    const int rowTile = bm * 64 + (wave >> 1) * 16;
    const int colTile = bn * 32 + (wave & 1) * 16;
    const int col = colTile + p;

    const float* __restrict__ Arow = X + (size_t)(rowTile + p) * H;

    v8f acc = {};
    #pragma unroll 4
    for (int k0 = 0; k0 < H; k0 += 4) {
        const int ka = k0 + 2 * hi;
        float a0 = Arow[ka];
        float a1 = Arow[ka + 1];
        if (APPLY_BN) {
            a0 = fmaxf(fmaf(a0, s_sc[ka],     s_sh[ka]),     0.0f);
            a1 = fmaxf(fmaf(a1, s_sc[ka + 1], s_sh[ka + 1]), 0.0f);
        }
        v2f a = {a0, a1};
        v2f b = {W[(size_t)ka * H + col], W[(size_t)(ka + 1) * H + col]};
        acc = __builtin_amdgcn_wmma_f32_16x16x4_f32(
            /*neg_a=*/false, a, /*neg_b=*/false, b,
            /*c_mod=*/(short)0, acc, /*reuse_a=*/false, /*reuse_b=*/false);
    }

    const float bb = bias[col];
    #pragma unroll
    for (int i = 0; i < 8; ++i)
        Y[(size_t)(rowTile + i + 8 * hi) * H + col] = acc[i] + bb;
}

// ---------------------------------------------------------------------------
// 3a) Column stats, stage 1: 32 blocks x 256 threads, each block reduces 128
//     rows (coalesced) into per-column partial sum / sum-of-squares.
// ---------------------------------------------------------------------------
__global__ __launch_bounds__(256)
void k_bn_partial(const float* __restrict__ Y,
                  float* __restrict__ psum,
                  float* __restrict__ psq) {
    const int blk = blockIdx.x;          // 0..31
    const int t   = threadIdx.x;         // column
    const float* base = Y + (size_t)blk * 128 * H;
    float s = 0.0f, q = 0.0f;
    #pragma unroll 8
    for (int r = 0; r < 128; ++r) {
        float v = base[(size_t)r * H + t];
        s += v;
        q += v * v;
    }
    psum[blk * H + t] = s;
    psq [blk * H + t] = q;
}

// ---------------------------------------------------------------------------
// 3b) Column stats, stage 2: fold partials, produce per-column BN scale/shift:
//     scale = gamma * rsqrt(var+eps),  shift = beta - scale*mean.
// ---------------------------------------------------------------------------
__global__ __launch_bounds__(256)
void k_bn_final(const float* __restrict__ psum,
                const float* __restrict__ psq,
                const float* __restrict__ gamma,
                const float* __restrict__ beta,
                float* __restrict__ ss) {
    const int c = threadIdx.x;           // 256 threads, 1 block
    float s = 0.0f, q = 0.0f;
    #pragma unroll
    for (int i = 0; i < 32; ++i) {
        s += psum[i * H + c];
        q += psq [i * H + c];
    }
    const float mean = s * (1.0f / (float)B_SAMP);
    const float var  = q * (1.0f / (float)B_SAMP) - mean * mean;
    const float inv  = rsqrtf(var + BN_EPS);
    const float sc   = gamma[c] * inv;
    ss[c]     = sc;
    ss[H + c] = beta[c] - sc * mean;
}

// ---------------------------------------------------------------------------
// 4) Output head: z[row] = relu(bn(y3[row])) . Wo + bo.  One block per row.
// ---------------------------------------------------------------------------
__global__ __launch_bounds__(256)
void k_out_dot(const float* __restrict__ Y3,
               const float* __restrict__ ss,
               const float* __restrict__ Wo,
               const float* __restrict__ bo,
               float* __restrict__ z) {
    const int row = blockIdx.x;
    const int t   = threadIdx.x;         // 256
    float v = Y3[(size_t)row * H + t];
    v = fmaxf(fmaf(v, ss[t], ss[H + t]), 0.0f);
    __shared__ float red[H];
    red[t] = v * Wo[t];
    __syncthreads();
    #pragma unroll
    for (int s = 128; s > 0; s >>= 1) {
        if (t < s) red[t] += red[t + s];
        __syncthreads();
    }
    if (t == 0) z[row] = red[0] + bo[0];
}

// ---------------------------------------------------------------------------
// 5) BCEWithLogits mean loss over B samples (stable form).
// ---------------------------------------------------------------------------
__global__ __launch_bounds__(1024)
void k_loss(const float* __restrict__ z,
            const float* __restrict__ tgt,
            float* __restrict__ out_loss) {
    const int i = threadIdx.x;           // 1024 threads, 1 block
    float s = 0.0f;
    for (int j = i; j < B_SAMP; j += 1024) {
        const float zz = z[j];
        const float tt = tgt[j];
        s += fmaxf(zz, 0.0f) - zz * tt + log1pf(expf(-fabsf(zz)));
    }
    __shared__ float red[1024];
    red[i] = s;
    __syncthreads();
    #pragma unroll
    for (int k = 512; k > 0; k >>= 1) {
        if (i < k) red[i] += red[i + k];
        __syncthreads();
    }
    if (i == 0) out_loss[0] = red[0] * (1.0f / (float)B_SAMP);
}

// ---------------------------------------------------------------------------
// Launch. Workspace layout (floats):
//   buf0 [0, 1M)      buf1 [1M, 2M)      psum [2M, 2M+8K)
//   psq  [2M+8K, 2M+16K)                 ss   [2M+16K, 2M+16K+512)
// Activations ping-pong buf0/buf1. Total ~8.5 MB.  No atomics anywhere ->
// bitwise-deterministic output across replays.
// ---------------------------------------------------------------------------
extern "C" void kernel_launch(void* const* d_in, const int* in_sizes, int n_in,
                              void* d_out, int out_size, void* d_ws, size_t ws_size,
                              hipStream_t stream) {
    const int*   tok = (const int*)  d_in[0];
    // d_in[1] = segment_ids (structural: repeat(arange(B), L); unused directly)
    const float* tgt = (const float*)d_in[2];
    const float* emb = (const float*)d_in[3];
    const float* W1  = (const float*)d_in[4];
    const float* b1  = (const float*)d_in[5];
    const float* g1  = (const float*)d_in[6];
    const float* be1 = (const float*)d_in[7];
    const float* W2  = (const float*)d_in[8];
    const float* b2  = (const float*)d_in[9];
    const float* g2  = (const float*)d_in[10];
    const float* be2 = (const float*)d_in[11];
    const float* W3  = (const float*)d_in[12];
    const float* b3  = (const float*)d_in[13];
    const float* g3  = (const float*)d_in[14];
    const float* be3 = (const float*)d_in[15];
    const float* Wo  = (const float*)d_in[16];
    const float* bo  = (const float*)d_in[17];

    float* ws   = (float*)d_ws;
    float* buf0 = ws;                         // 4096*256
    float* buf1 = ws + (size_t)B_SAMP * H;    // 4096*256
    float* psum = buf1 + (size_t)B_SAMP * H;  // 32*256
    float* psq  = psum + 32 * H;              // 32*256
    float* ss   = psq  + 32 * H;              // 512

    float* out  = (float*)d_out;              // [loss, z(4096)]
    float* z    = out + 1;

    const dim3 blk256(256);
    const int gemmGrid = (B_SAMP / 64) * (H / 32);   // 512

    // bow = segment-mean of gathered embeddings
    k_gather_mean<<<B_SAMP, blk256, 0, stream>>>(tok, emb, buf0);

    // Layer 1: y1 = bow @ W1 + b1 ; stats(y1) with (g1,be1)
    k_gemm_bn<0><<<gemmGrid, blk256, 0, stream>>>(buf0, W1, b1, buf1, ss);
    k_bn_partial<<<32, blk256, 0, stream>>>(buf1, psum, psq);
    k_bn_final<<<1, blk256, 0, stream>>>(psum, psq, g1, be1, ss);

    // Layer 2: y2 = bnrelu(y1) @ W2 + b2 ; stats(y2) with (g2,be2)
    k_gemm_bn<1><<<gemmGrid, blk256, 0, stream>>>(buf1, W2, b2, buf0, ss);
    k_bn_partial<<<32, blk256, 0, stream>>>(buf0, psum, psq);
    k_bn_final<<<1, blk256, 0, stream>>>(psum, psq, g2, be2, ss);

    // Layer 3: y3 = bnrelu(y2) @ W3 + b3 ; stats(y3) with (g3,be3)
    k_gemm_bn<1><<<gemmGrid, blk256, 0, stream>>>(buf0, W3, b3, buf1, ss);
    k_bn_partial<<<32, blk256, 0, stream>>>(buf1, psum, psq);
    k_bn_final<<<1, blk256, 0, stream>>>(psum, psq, g3, be3, ss);

    // Head: z = bnrelu(y3) @ Wo + bo ; then BCE mean loss
    k_out_dot<<<B_SAMP, blk256, 0, stream>>>(buf1, ss, Wo, bo, z);
    k_loss<<<1, dim3(1024), 0, stream>>>(z, tgt, out);
}